// model_52922587021824
// MI455X (gfx1250) — compile-verified
//
#include <hip/hip_runtime.h>

typedef float v2f __attribute__((ext_vector_type(2)));
typedef float v8f __attribute__((ext_vector_type(8)));

#define K_   128
#define CS_  8
#define NS_  20
#define LOG_SQRT_2PI_D 0.9189385332046727

__device__ __forceinline__ float softplus_f(float x) {
    // numerically stable softplus
    return fmaxf(x, 0.0f) + log1pf(expf(-fabsf(x)));
}

__global__ void init_ws_kernel(float* ws) {
    ws[0] = 0.0f;  // log-likelihood accumulator
    ws[1] = 0.0f;  // sum-of-squares accumulator
}

// One wave handles 16 batch rows. ctx_sum fragment lives in registers in
// exactly the A-matrix layout for V_WMMA_F32_16X16X4_F32; per-target B
// fragments are gathered straight from target_W in B-matrix layout.
__global__ __launch_bounds__(256) void sgns_main_kernel(
    const int*   __restrict__ contexts,   // (B, CS)
    const int*   __restrict__ targets,    // (B, 1)
    const int*   __restrict__ neg_idx,    // (B, NS)
    const float* __restrict__ target_W,   // (L, K)
    const float* __restrict__ context_W,  // (L+1, K)
    float*       __restrict__ ws)
{
    const int ln   = threadIdx.x & 31;
    const int wave = threadIdx.x >> 5;
    const int tile = blockIdx.x * 8 + wave;   // 16 batch rows per wave
    const int b0   = tile * 16;
    const int r    = ln & 15;                 // batch row within tile
    const int h    = ln >> 4;                 // half: cols {2h,2h+1} of each K=4 chunk

    // --- ctx_sum fragment: ctx[c] = ctx_sum[r][4c+2h .. 4c+2h+1] ---
    v2f ctx[32] = {};
    const int* cp = contexts + (size_t)(b0 + r) * CS_;
#pragma unroll
    for (int s = 0; s < CS_; ++s) {
        const v2f* wp = (const v2f*)(context_W + (size_t)cp[s] * K_) + h;
#pragma unroll
        for (int c = 0; c < 32; ++c) ctx[c] += wp[2 * c];
    }

    // diagonal ownership: m<8 -> lane m holds acc[m]; m>=8 -> lane m+16 holds acc[m-8]
    const bool owner = (ln < 8) || (ln >= 24);
    const int  sel   = ((ln < 8) ? ln : (ln - 24)) & 7;

    float lsum = 0.0f;

#pragma unroll 1
    for (int j = 0; j <= NS_; ++j) {
        const int tIdx = (j == 0) ? targets[b0 + r]
                                  : neg_idx[(size_t)(b0 + r) * NS_ + (j - 1)];
        const v2f* tp = (const v2f*)(target_W + (size_t)tIdx * K_) + h;

        v8f acc = {};
#pragma unroll
        for (int c = 0; c < 32; ++c) {
            // D = A(16x4,f32) * B(4x16,f32) + C ; fp32 WMMA, K accumulated 4 at a time
            acc = __builtin_amdgcn_wmma_f32_16x16x4_f32(
                false, ctx[c], false, tp[2 * c], (short)0, acc, false, false);
        }

        // branchless diagonal extraction (EXEC stays all-ones around WMMAs)
        float diag = acc[0];
        diag = (sel == 1) ? acc[1] : diag;
        diag = (sel == 2) ? acc[2] : diag;
        diag = (sel == 3) ? acc[3] : diag;
        diag = (sel == 4) ? acc[4] : diag;
        diag = (sel == 5) ? acc[5] : diag;
        diag = (sel == 6) ? acc[6] : diag;
        diag = (sel == 7) ? acc[7] : diag;

        const float x = (j == 0) ? -diag : diag;   // pos: -sp(-eta), neg: -sp(eta)
        lsum += owner ? -softplus_f(x) : 0.0f;
    }

    // wave32 reduction + single atomic per wave
#pragma unroll
    for (int off = 16; off > 0; off >>= 1)
        lsum += __shfl_xor(lsum, off, 32);
    if (ln == 0) atomicAdd(&ws[0], lsum);
}

// Streaming sum of squares over both weight tables (float4 loads).
__global__ __launch_bounds__(256) void prior_sumsq_kernel(
    const float* __restrict__ target_W, long long n4_t,
    const float* __restrict__ context_W, long long n4_c,
    float* __restrict__ ws)
{
    const long long n4     = n4_t + n4_c;
    const long long stride = (long long)gridDim.x * blockDim.x;
    float s = 0.0f;
    for (long long i = (long long)blockIdx.x * blockDim.x + threadIdx.x;
         i < n4; i += stride) {
        const float4* p;
        long long k;
        if (i < n4_t) { p = (const float4*)target_W;  k = i; }
        else          { p = (const float4*)context_W; k = i - n4_t; }
        float4 v = p[k];
        s += v.x * v.x + v.y * v.y + v.z * v.z + v.w * v.w;
    }
#pragma unroll
    for (int off = 16; off > 0; off >>= 1)
        s += __shfl_xor(s, off, 32);
    if ((threadIdx.x & 31) == 0) atomicAdd(&ws[1], s);
}

__global__ void finalize_kernel(const float* ws, float* out, long long count) {
    const double ll   = (double)ws[0];
    const double ssq  = (double)ws[1];
    const double logp = -0.5 * ssq - (double)count * LOG_SQRT_2PI_D;
    out[0] = (float)(-(10.0 * ll + logp));
}

extern "C" void kernel_launch(void* const* d_in, const int* in_sizes, int n_in,
                              void* d_out, int out_size, void* d_ws, size_t ws_size,
                              hipStream_t stream) {
    (void)n_in; (void)out_size; (void)ws_size;
    const int*   contexts  = (const int*)  d_in[0];
    const int*   targets   = (const int*)  d_in[1];
    const int*   neg_idx   = (const int*)  d_in[2];
    const float* target_W  = (const float*)d_in[3];
    const float* context_W = (const float*)d_in[4];
    float* out = (float*)d_out;
    float* ws  = (float*)d_ws;

    const int B = in_sizes[1];                       // 16384
    const long long n_t = in_sizes[3];               // L*K
    const long long n_c = in_sizes[4];               // (L+1)*K
    const long long count = n_t + n_c;

    init_ws_kernel<<<1, 1, 0, stream>>>(ws);

    // 8 waves/block, 16 rows/wave -> 128 rows/block
    sgns_main_kernel<<<B / 128, 256, 0, stream>>>(
        contexts, targets, neg_idx, target_W, context_W, ws);

    prior_sumsq_kernel<<<1024, 256, 0, stream>>>(
        target_W, n_t / 4, context_W, n_c / 4, ws);

    finalize_kernel<<<1, 1, 0, stream>>>(ws, out, count);
}